// ScaledDotProductAttention_7619271983556
// MI455X (gfx1250) — compile-verified
//
#include <hip/hip_runtime.h>
#include <hip/hip_bf16.h>

#define SZ_B     64
#define SEQ_LEN  1024
#define N_HEADS  16
#define D_K      64
#define EMB      64
#define N_TOTAL  (SZ_B * N_HEADS)   // 1024

typedef float v2f __attribute__((ext_vector_type(2)));
typedef float v8f __attribute__((ext_vector_type(8)));

// -----------------------------------------------------------------------------
// Kernel 1: per-n scores via V_WMMA_F32_16X16X4_F32, then softmax over seq.
// Grid: N_TOTAL blocks of 256 threads (8 waves). Each wave owns 128 rows of K.
//
// WMMA trick: A = q-chunk broadcast over all 16 M rows, B = K^T tile, so
// D[m,j] = sum_k q[k] * K[s0+j,k] for every m. D VGPR0 holds score[j] in lane j
// (C/D layout: VGPR0, lanes 0..15 = M=0, N=0..15).
// A/B f32 layouts: lane holds k = {0,1}+2*(lane>=16) within each 4-wide chunk,
// i.e. one contiguous float2 per lane per step.
// -----------------------------------------------------------------------------
__global__ void __launch_bounds__(256)
attn_scores_softmax_kernel(const float* __restrict__ q,
                           const float* __restrict__ k,
                           const unsigned char* __restrict__ mask,
                           float* __restrict__ attn_out) {
  __shared__ float sc[SEQ_LEN];
  __shared__ float qs[D_K];
  __shared__ float red[8];

  const int n    = blockIdx.x;
  const int tid  = threadIdx.x;
  const int lane = tid & 31;
  const int wave = tid >> 5;

  // Stage q[n,:] (256B) into LDS once; reused by every WMMA step.
  if (tid < D_K) qs[tid] = q[(size_t)n * D_K + tid];
  __syncthreads();

  const int koffLane = (lane >> 4) << 1;  // lanes 0-15 -> +0, lanes 16-31 -> +2
  const int col      = lane & 15;         // B-matrix column (seq index within tile)
  const size_t kbase = (size_t)n * SEQ_LEN * D_K;

  // 8 waves x 8 tiles x 16 seq-rows = 1024 scores
  for (int t = 0; t < 8; ++t) {
    const int s0 = wave * 128 + t * 16;
    const float* krow = k + kbase + (size_t)(s0 + col) * D_K;

    v8f c = {};
#pragma unroll
    for (int kc = 0; kc < 16; ++kc) {
      const int ko = kc * 4 + koffLane;
      v2f a, b;
      a[0] = qs[ko];     a[1] = qs[ko + 1];      // A: q broadcast over rows
      b[0] = krow[ko];   b[1] = krow[ko + 1];    // B: K^T tile, contiguous 8B
      c = __builtin_amdgcn_wmma_f32_16x16x4_f32(
          /*neg_a=*/false, a, /*neg_b=*/false, b,
          /*c_mod=*/(short)0, c, /*reuse_a=*/false, /*reuse_b=*/false);
    }
    // score[j] for j = s0..s0+15 sits in lane j-s0 of c[0]
    if (lane < 16) {
      const int j = s0 + lane;
      float s = c[0] * 0.125f;                   // / TEMPERATURE
      if (mask[(size_t)n * SEQ_LEN + j]) s = -1.0e8f;
      sc[j] = s;
    }
  }
  __syncthreads();

  // ---- softmax over 1024 values in LDS (256 threads, 4 values each) ----
  float vals[4];
  float m = -3.4e38f;
#pragma unroll
  for (int i = 0; i < 4; ++i) {
    vals[i] = sc[tid + i * 256];
    m = fmaxf(m, vals[i]);
  }
#pragma unroll
  for (int off = 16; off > 0; off >>= 1) m = fmaxf(m, __shfl_xor(m, off));
  if (lane == 0) red[wave] = m;
  __syncthreads();
  if (tid == 0) {
    float mm = red[0];
    for (int i = 1; i < 8; ++i) mm = fmaxf(mm, red[i]);
    red[0] = mm;
  }
  __syncthreads();
  m = red[0];

  float ssum = 0.f;
#pragma unroll
  for (int i = 0; i < 4; ++i) {
    vals[i] = __expf(vals[i] - m);
    ssum += vals[i];
  }
#pragma unroll
  for (int off = 16; off > 0; off >>= 1) ssum += __shfl_xor(ssum, off);
  __syncthreads();                 // red[] reuse hazard
  if (lane == 0) red[wave] = ssum;
  __syncthreads();
  if (tid == 0) {
    float t2 = 0.f;
    for (int i = 0; i < 8; ++i) t2 += red[i];
    red[0] = t2;
  }
  __syncthreads();
  const float inv = 1.0f / red[0];

  // Write attn in reference output layout [b, s, 1, h]; region is L2-resident.
  const int b = n >> 4, h = n & 15;
#pragma unroll
  for (int i = 0; i < 4; ++i) {
    const int j = tid + i * 256;
    attn_out[((size_t)b * SEQ_LEN + j) * N_HEADS + h] = vals[i] * inv;
  }
}

// -----------------------------------------------------------------------------
// Kernel 2: out[b,s,e] = sum_h attn[b,s,h] * v[b,s,h,e]
// Batched (1x16)x(16x64) per (b,s) -> VALU FMA chain, V streamed once,
// contiguous 256B per h per 64-lane group. 256 threads handle 4 (b,s) pairs.
// -----------------------------------------------------------------------------
__global__ void __launch_bounds__(256)
attn_apply_v_kernel(const float* __restrict__ attn,
                    const float* __restrict__ v,
                    float* __restrict__ out) {
  const int tid  = threadIdx.x;
  const int pair = blockIdx.x * 4 + (tid >> 6);  // flat (b*SEQ_LEN + s)
  const int e    = tid & 63;

  const float* a  = attn + (size_t)pair * N_HEADS;
  const float* vp = v + (size_t)pair * N_HEADS * EMB + e;

  float acc = 0.f;
#pragma unroll
  for (int h = 0; h < N_HEADS; ++h) acc = fmaf(a[h], vp[h * EMB], acc);

  out[(size_t)pair * EMB + e] = acc;
}

extern "C" void kernel_launch(void* const* d_in, const int* in_sizes, int n_in,
                              void* d_out, int out_size, void* d_ws, size_t ws_size,
                              hipStream_t stream) {
  const float*         q    = (const float*)d_in[0];          // [N, 64]
  const float*         k    = (const float*)d_in[1];          // [N, 1024, 64]
  const float*         v    = (const float*)d_in[2];          // [64, 1024, 16, 64]
  const unsigned char* mask = (const unsigned char*)d_in[3];  // [N, 1, 1024] bool

  float* out  = (float*)d_out;                       // [64, 1024, 64]
  float* attn = out + (size_t)SZ_B * SEQ_LEN * EMB;  // [64, 1024, 1, 16]

  attn_scores_softmax_kernel<<<N_TOTAL, 256, 0, stream>>>(q, k, mask, attn);
  attn_apply_v_kernel<<<(SZ_B * SEQ_LEN) / 4, 256, 0, stream>>>(attn, v, out);
}